// OverfitResonanceModel_25323127177675
// MI455X (gfx1250) — compile-verified
//
#include <hip/hip_runtime.h>
#include <hip/hip_bf16.h>

// ---------------------------------------------------------------------------
// OverfitResonanceModel for MI455X (gfx1250).
//  * sparse_softmax selections  -> argmax row gathers (tiny prep kernel)
//  * rsel @ waves (128x4096 x 4096x32768 fp32 GEMM) -> v_wmma_f32_16x16x4_f32,
//    A tile (16x4096 = 256KB) staged in LDS (320KB/WGP), B streamed from HBM
//    (arith intensity ~64 FLOP/B vs 23.3 TB/s HBM -> BW-balanced, ~22us floor).
//  * 512-tap FIR convs -> direct LDS-tiled time domain (~3 GMAC total)
//  * full 32768-length convs -> 32768-pt complex FFT in LDS per workgroup
//    (even/odd real packing of the 65536-pt zero-padded transform; 256KB LDS),
//    spectra kept packed in HBM, unpack*mul*repack fused into inverse kernel.
//  * A-tile staging + FFT bit-reversed loads / spectra stores use the CDNA5
//    async global<->LDS path (ASYNCcnt) when the toolchain exposes it.
// Workspace requirement: ~110 MB floats (see layout in kernel_launch).
// ---------------------------------------------------------------------------

#define E        32
#define S        32768
#define FRAMES   128
#define HOP      256
#define IX       4
#define NX       2
#define NFS      512
#define ENV_SAMPS 128
#define ENV_FULL 8192
#define FFT_N    32768
#define FFT_LOG2 15

// ---- CDNA5 async global<->LDS (ASYNCcnt) feature detection ----------------
#if defined(__has_builtin)
#if __has_builtin(__builtin_amdgcn_global_load_async_to_lds_b32) && \
    __has_builtin(__builtin_amdgcn_global_load_async_to_lds_b64) && \
    __has_builtin(__builtin_amdgcn_global_store_async_from_lds_b64) && \
    __has_builtin(__builtin_amdgcn_s_wait_asynccnt)
#define USE_ASYNC_LDS 1
#endif
#endif
#ifndef USE_ASYNC_LDS
#define USE_ASYNC_LDS 0
#endif

#if USE_ASYNC_LDS
// Builtin params are typed, address-space-qualified pointers:
//  b32: (as1 int*, as3 int*, imm, imm);  b64: (as1 int2*, as3 int2*, imm, imm)
typedef int v2i_async __attribute__((ext_vector_type(2)));
#define GAS_I32(p) ((__attribute__((address_space(1))) int*)(p))
#define LAS_I32(p) ((__attribute__((address_space(3))) int*)(p))
#define GAS_I64(p) ((__attribute__((address_space(1))) v2i_async*)(p))
#define LAS_I64(p) ((__attribute__((address_space(3))) v2i_async*)(p))
#endif

typedef __attribute__((ext_vector_type(2))) float v2f;
typedef __attribute__((ext_vector_type(8))) float v8f;

struct cx { float x, y; };

__device__ __forceinline__ cx cmul(cx a, cx b) {
  return { a.x * b.x - a.y * b.y, a.x * b.y + a.y * b.x };
}

__device__ __forceinline__ int dev_argmax(const float* p, int n) {
  int bi = 0; float bv = p[0];
  for (int i = 1; i < n; ++i) { if (p[i] > bv) { bv = p[i]; bi = i; } }
  return bi;
}

__device__ __forceinline__ void softmax2(const float* in, float* out) {
  float m = fmaxf(in[0], in[1]);
  float a = __expf(in[0] - m), b = __expf(in[1] - m);
  float s = a + b;
  out[0] = a / s; out[1] = b / s;
}

// jax.image.resize 'linear' (half-pixel centers, clamped edges)
__device__ __forceinline__ float interp_row(const float* row, int n_in, float scale, int t) {
  float c = ((float)t + 0.5f) * scale - 0.5f;
  int i0 = (int)floorf(c);
  float fr = c - (float)i0;
  int ia = i0;     if (ia < 0) ia = 0; if (ia > n_in - 1) ia = n_in - 1;
  int ib = i0 + 1; if (ib < 0) ib = 0; if (ib > n_in - 1) ib = n_in - 1;
  float a = row[ia], b = row[ib];
  return a + (b - a) * fr;
}

// ---------------------------------------------------------------------------
// Prep: argmax selections, 2-way softmax mixes, deformation frame softmaxes,
// decay cumprod, hamming-windowed FIR tables.  One block, 256 threads.
// ---------------------------------------------------------------------------
__global__ void k_prep(
    const float* __restrict__ envelopes, const float* __restrict__ resonances,
    const float* __restrict__ noise_resonance, const float* __restrict__ res_filter,
    const float* __restrict__ deformations, const float* __restrict__ noise_deformations,
    const float* __restrict__ decays, const float* __restrict__ times,
    const float* __restrict__ room_choice, const float* __restrict__ noise_mixes,
    const float* __restrict__ mixes, const float* __restrict__ room_mix,
    const float* __restrict__ d_items, const float* __restrict__ warp_items,
    const float* __restrict__ noise_warp_items, const float* __restrict__ n_items,
    int* sel_env, int* sel_res, int* sel_nres, int* sel_rf, int* sel_def,
    int* sel_ndef, int* sel_dec, int* sel_time, int* sel_room,
    float* nmix, float* mmix, float* rmix2, float* defoF, float* ndefF,
    float* dec128, float* rfF, float* nF)
{
  const int tid = threadIdx.x;
  if (tid < E) {
    int e = tid;
    sel_env[e]  = dev_argmax(envelopes + e * 64, 64);
    sel_def[e]  = dev_argmax(deformations + e * 32, 32);
    sel_ndef[e] = dev_argmax(noise_deformations + e * 16, 16);
    sel_dec[e]  = dev_argmax(decays + e * 64, 64);
    sel_time[e] = dev_argmax(times + e * FRAMES, FRAMES);
    sel_room[e] = dev_argmax(room_choice + e * 8, 8);
    softmax2(noise_mixes + e * 2, nmix + e * 2);
    softmax2(mixes + e * 2, mmix + e * 2);
    softmax2(room_mix + e * 2, rmix2 + e * 2);
  }
  for (int i = tid; i < E * IX; i += 256) sel_res[i] = dev_argmax(resonances + i * 64, 64);
  for (int i = tid; i < E * NX; i += 256) {
    sel_nres[i] = dev_argmax(noise_resonance + i * 32, 32);
    sel_rf[i]   = dev_argmax(res_filter + i * 32, 32);
  }
  __syncthreads();

  // decay cumprod (per e, sequential over 128 frames)
  if (tid < E) {
    int e = tid;
    const float* dr = d_items + sel_dec[e] * FRAMES;
    float lacc = 0.0f;
    for (int f2 = 0; f2 < FRAMES; ++f2) {
      float dp = 0.5f + 0.5f / (1.0f + __expf(-dr[f2]));
      lacc += __logf(dp + 1e-12f);
      dec128[e * FRAMES + f2] = __expf(lacc);
    }
  }
  // deformation frames: softmax over channel axis at each frame
  for (int i = tid; i < E * FRAMES; i += 256) {
    int e = i / FRAMES, f2 = i % FRAMES;
    const float* row = warp_items + sel_def[e] * (IX * FRAMES);
    float v[IX], mx = -1e30f;
    for (int x = 0; x < IX; ++x) { v[x] = row[x * FRAMES + f2]; mx = fmaxf(mx, v[x]); }
    float s = 0.0f;
    for (int x = 0; x < IX; ++x) { v[x] = __expf(v[x] - mx); s += v[x]; }
    for (int x = 0; x < IX; ++x) defoF[(e * IX + x) * FRAMES + f2] = v[x] / s;
  }
  for (int i = tid; i < E * FRAMES; i += 256) {
    int e = i / FRAMES, f2 = i % FRAMES;
    const float* row = noise_warp_items + sel_ndef[e] * (NX * FRAMES);
    float v0 = row[f2], v1 = row[FRAMES + f2];
    float mx = fmaxf(v0, v1);
    float a = __expf(v0 - mx), b = __expf(v1 - mx);
    float s = a + b;
    ndefF[(e * NX + 0) * FRAMES + f2] = a / s;
    ndefF[(e * NX + 1) * FRAMES + f2] = b / s;
  }
  // hamming-windowed FIR tables
  for (int i = tid; i < E * NFS; i += 256) {
    int e = i / NFS, t = i % NFS;
    float ham = 0.54f - 0.46f * __cosf(6.28318530717959f * (float)t / (float)NFS);
    rfF[i] = (n_items[sel_rf[e * 2 + 0] * NFS + t] +
              n_items[sel_rf[e * 2 + 1] * NFS + t]) * ham;
  }
  for (int i = tid; i < E * NX * NFS; i += 256) {
    int ex = i / NFS, t = i % NFS;
    float ham = 0.54f - 0.46f * __cosf(6.28318530717959f * (float)t / (float)NFS);
    nF[i] = n_items[sel_nres[ex] * NFS + t] * ham;
  }
}

// ---------------------------------------------------------------------------
// Envelope: select e_items row, square, interp 128->8192, * env_noise, pad to S
// ---------------------------------------------------------------------------
__global__ void k_env(const float* __restrict__ e_items, const float* __restrict__ env_noise,
                      const int* __restrict__ sel_env, float* __restrict__ impulses)
{
  int e = blockIdx.y;
  int t = blockIdx.x * 256 + threadIdx.x;
  float v = 0.0f;
  if (t < ENV_FULL) {
    const float* row = e_items + sel_env[e] * ENV_SAMPS;
    float c = ((float)t + 0.5f) * ((float)ENV_SAMPS / (float)ENV_FULL) - 0.5f;
    int i0 = (int)floorf(c);
    float fr = c - (float)i0;
    int ia = i0;     if (ia < 0) ia = 0; if (ia > ENV_SAMPS - 1) ia = ENV_SAMPS - 1;
    int ib = i0 + 1; if (ib < 0) ib = 0; if (ib > ENV_SAMPS - 1) ib = ENV_SAMPS - 1;
    float a = row[ia]; a = a * a;   // e_items ** 2 (one-hot select commutes)
    float b = row[ib]; b = b * b;
    v = (a + (b - a) * fr) * env_noise[(size_t)e * ENV_FULL + t];
  }
  impulses[(size_t)e * S + t] = v;
}

// Gather A = r_items[sel_res]  (128 x 4096)
__global__ void k_build_A(const float* __restrict__ r_items, const int* __restrict__ sel_res,
                          float* __restrict__ A)
{
  int m = blockIdx.y;
  int k = blockIdx.x * 256 + threadIdx.x;
  A[(size_t)m * 4096 + k] = r_items[(size_t)sel_res[m] * 4096 + k];
}

// ---------------------------------------------------------------------------
// WMMA GEMM: res0(128 x 32768) = A(128 x 4096) * waves(4096 x 32768), fp32.
// 256 threads = 8 waves; each wave owns one 16x16 output tile; block shares a
// 16-row A tile resident in LDS (256 KB) in pair-K layout so each lane's
// {K,K+1} fragment is a single 8-byte ds load.  A staging uses the async
// global->LDS scatter path when available.
// ---------------------------------------------------------------------------
__global__ __launch_bounds__(256) void k_wave_gemm(
    const float* __restrict__ A, const float* __restrict__ waves, float* __restrict__ res0)
{
  extern __shared__ float ldsA[];      // 16*4096 floats, layout [k/2][m][k&1]
  const int tid = threadIdx.x;
  const int mtile = blockIdx.y;

#if USE_ASYNC_LDS
  for (int i = tid; i < 16 * 4096; i += 256) {
    int m = i & 15, k = i >> 4;
    __builtin_amdgcn_global_load_async_to_lds_b32(
        GAS_I32(&A[(size_t)(mtile * 16 + m) * 4096 + k]),
        LAS_I32(&ldsA[(k >> 1) * 32 + m * 2 + (k & 1)]), 0, 0);
  }
  __builtin_amdgcn_s_wait_asynccnt(0);
#else
  for (int i = tid; i < 16 * 4096; i += 256) {
    int m = i & 15, k = i >> 4;
    ldsA[(k >> 1) * 32 + m * 2 + (k & 1)] =
        A[(size_t)(mtile * 16 + m) * 4096 + k];
  }
#endif
  __syncthreads();

  const int wave = tid >> 5, lane = tid & 31;
  const int ntile = blockIdx.x * 8 + wave;
  const int n = ntile * 16 + (lane & 15);
  const int koff = (lane >> 4) * 2;          // lanes 0-15: K+0/1, lanes 16-31: K+2/3
  const int mlane = (lane & 15) * 2;

  v8f c = {};
#pragma unroll 4
  for (int k = 0; k < 4096; k += 4) {
    const int k0 = k + koff;
    const float* ap = &ldsA[(k0 >> 1) * 32 + mlane];
    v2f av = { ap[0], ap[1] };
    v2f bv = { waves[(size_t)k0 * S + n], waves[(size_t)(k0 + 1) * S + n] };
    c = __builtin_amdgcn_wmma_f32_16x16x4_f32(
        /*neg_a=*/false, av, /*neg_b=*/false, bv,
        /*c_mod=*/(short)0, c, /*reuse_a=*/false, /*reuse_b=*/false);
  }

  const int rowbase = (lane >> 4) * 8;       // C/D layout: lanes 16-31 hold M=8..15
#pragma unroll
  for (int j = 0; j < 8; ++j) {
    int m = mtile * 16 + rowbase + j;
    res0[(size_t)m * S + n] = c[j];
  }
}

// ---------------------------------------------------------------------------
// Noise stage: imp = impulses*nm0 + (sum_x conv512(impulses, nF_x)*ndef_x)*nm1
// ---------------------------------------------------------------------------
__global__ void k_noise(const float* __restrict__ impulses, const float* __restrict__ nF,
                        const float* __restrict__ ndefF, const float* __restrict__ nmix,
                        float* __restrict__ imp)
{
  __shared__ float sImp[768];
  __shared__ float sF[2 * NFS];
  const int e = blockIdx.y;
  const int t0 = blockIdx.x * 256;
  const int tid = threadIdx.x;

  for (int i = tid; i < 767; i += 256) {
    int g = t0 - (NFS - 1) + i;
    sImp[i] = (g >= 0) ? impulses[(size_t)e * S + g] : 0.0f;
  }
  for (int i = tid; i < 2 * NFS; i += 256) sF[i] = nF[(size_t)e * 2 * NFS + i];
  __syncthreads();

  float acc0 = 0.0f, acc1 = 0.0f;
#pragma unroll 8
  for (int tau = 0; tau < NFS; ++tau) {
    float xv = sImp[tid + (NFS - 1) - tau];
    acc0 += sF[tau] * xv;
    acc1 += sF[NFS + tau] * xv;
  }
  const int t = t0 + tid;
  const float sc = (float)FRAMES / (float)S;
  float w0 = interp_row(ndefF + (e * NX + 0) * FRAMES, FRAMES, sc, t);
  float w1 = interp_row(ndefF + (e * NX + 1) * FRAMES, FRAMES, sc, t);
  float wet = acc0 * w0 + acc1 * w1;
  imp[(size_t)e * S + t] = impulses[(size_t)e * S + t] * nmix[e * 2] + wet * nmix[e * 2 + 1];
}

// res = conv512(res0, rfF[e]) * interp(dec128[e])   (128 signals)
__global__ void k_resconv(const float* __restrict__ res0, const float* __restrict__ rfF,
                          const float* __restrict__ dec128, float* __restrict__ res)
{
  __shared__ float sImp[768];
  __shared__ float sF[NFS];
  const int sig = blockIdx.y;
  const int e = sig >> 2;
  const int t0 = blockIdx.x * 256;
  const int tid = threadIdx.x;

  for (int i = tid; i < 767; i += 256) {
    int g = t0 - (NFS - 1) + i;
    sImp[i] = (g >= 0) ? res0[(size_t)sig * S + g] : 0.0f;
  }
  for (int i = tid; i < NFS; i += 256) sF[i] = rfF[(size_t)e * NFS + i];
  __syncthreads();

  float acc = 0.0f;
#pragma unroll 8
  for (int tau = 0; tau < NFS; ++tau) acc += sF[tau] * sImp[tid + (NFS - 1) - tau];

  const int t = t0 + tid;
  float d = interp_row(dec128 + e * FRAMES, FRAMES, (float)FRAMES / (float)S, t);
  res[(size_t)sig * S + t] = acc * d;
}

// ---------------------------------------------------------------------------
// In-LDS 32768-point complex FFT (radix-2 DIT, data pre-bit-reversed).
// 256 KB LDS per workgroup (fits the 320 KB/WGP CDNA5 budget), 1024 threads.
// ---------------------------------------------------------------------------
__device__ __forceinline__ void fft_lds(cx* z, float dir)
{
  const int tid = threadIdx.x;
  for (int s = 1; s <= FFT_LOG2; ++s) {
    const int mh = 1 << (s - 1);
    const float astep = dir * 3.14159265358979f / (float)mh;
    for (int b = tid; b < (FFT_N >> 1); b += 1024) {
      const int j = b & (mh - 1);
      const int i0 = ((b >> (s - 1)) << s) + j;
      const int i1 = i0 + mh;
      float sn, csn;
      __sincosf(astep * (float)j, &sn, &csn);
      cx u = z[i0], v = z[i1];
      cx t = { v.x * csn - v.y * sn, v.x * sn + v.y * csn };
      z[i0] = { u.x + t.x, u.y + t.y };
      z[i1] = { u.x - t.x, u.y - t.y };
    }
    __syncthreads();
  }
}

// Forward packed rFFT(2S) of a length-S real row (zero-padded to 2S):
// z[n] = x[2n] + i x[2n+1], stored bit-reversed, FFT in LDS, packed Z to HBM.
// Bit-reversal trick: for the 15-bit reversal r of j, j even <=> r < 16384
// <=> both packed samples are in-range (async b64 scatter); j odd <=> both
// are zero padding (plain LDS store).
__global__ __launch_bounds__(1024) void k_fft_fwd(
    const float* __restrict__ in, const int* __restrict__ rowIdx, cx* __restrict__ spec)
{
  extern __shared__ cx zf[];
  const int tid = threadIdx.x;
  const float* src = rowIdx ? (in + (size_t)rowIdx[blockIdx.x] * S)
                            : (in + (size_t)blockIdx.x * S);
#if USE_ASYNC_LDS
  for (int j = tid; j < FFT_N; j += 1024) {
    int r = __brev((unsigned)j) >> (32 - FFT_LOG2);
    if ((j & 1) == 0) {
      __builtin_amdgcn_global_load_async_to_lds_b64(
          GAS_I64(&src[2 * r]), LAS_I64(&zf[j]), 0, 0);
    } else {
      zf[j] = { 0.0f, 0.0f };
    }
  }
  __builtin_amdgcn_s_wait_asynccnt(0);
#else
  for (int j = tid; j < FFT_N; j += 1024) {
    int r = __brev((unsigned)j) >> (32 - FFT_LOG2);
    int t0 = r * 2;
    cx v;
    v.x = (t0 < S) ? src[t0] : 0.0f;
    v.y = (t0 + 1 < S) ? src[t0 + 1] : 0.0f;
    zf[j] = v;
  }
#endif
  __syncthreads();
  fft_lds(zf, -1.0f);
#if USE_ASYNC_LDS
  for (int j = tid; j < FFT_N; j += 1024) {
    __builtin_amdgcn_global_store_async_from_lds_b64(
        GAS_I64(&spec[(size_t)blockIdx.x * FFT_N + j]), LAS_I64(&zf[j]), 0, 0);
  }
  __builtin_amdgcn_s_wait_asynccnt(0);
#else
  for (int j = tid; j < FFT_N; j += 1024) spec[(size_t)blockIdx.x * FFT_N + j] = zf[j];
#endif
}

// Unpack packed Z to the true rFFT bin X[k] (N = FFT_N implicit).
__device__ __forceinline__ cx unpackX(cx Zk, cx Znk, float k)
{
  cx Ze = { 0.5f * (Zk.x + Znk.x),  0.5f * (Zk.y - Znk.y) };
  cx Zo = { 0.5f * (Zk.y + Znk.y), -0.5f * (Zk.x - Znk.x) };
  float sn, csn;
  __sincosf(-3.14159265358979f * k / (float)FFT_N, &sn, &csn);
  cx w = { csn, sn };
  cx wo = cmul(w, Zo);
  return { Ze.x + wo.x, Ze.y + wo.y };
}

// Fused: unpack both operand spectra, multiply, repack product into LDS
// (bit-reversed) ready for the inverse FFT.
__device__ void conv_spectra_to_lds(cx* z, const cx* __restrict__ A, const cx* __restrict__ B)
{
  const int tid = threadIdx.x;
  for (int k = tid; k <= FFT_N / 2; k += 1024) {
    const int nk = (FFT_N - k) & (FFT_N - 1);
    cx Ak = A[k], Bk = B[k], Ank = A[nk], Bnk = B[nk];
    cx Yk, Yp;                               // Y[k], Y[N-k]
    if (k == 0) {
      float Xa0 = Ak.x + Ak.y, XaN = Ak.x - Ak.y;
      float Xb0 = Bk.x + Bk.y, XbN = Bk.x - Bk.y;
      Yk = { Xa0 * Xb0, 0.0f };
      Yp = { XaN * XbN, 0.0f };              // Y[N] (real)
    } else {
      cx Xak  = unpackX(Ak, Ank, (float)k);
      cx Xbk  = unpackX(Bk, Bnk, (float)k);
      cx Xank = unpackX(Ank, Ak, (float)(FFT_N - k));
      cx Xbnk = unpackX(Bnk, Bk, (float)(FFT_N - k));
      Yk = cmul(Xak, Xbk);
      Yp = cmul(Xank, Xbnk);
    }
    // Zy[k] = E + iO,  E = (Y[k]+conj(Y[N-k]))/2,  O = e^{i pi k/N}(Y[k]-conj(Y[N-k]))/2
    cx Ep = { 0.5f * (Yk.x + Yp.x), 0.5f * (Yk.y - Yp.y) };
    cx Dp = { 0.5f * (Yk.x - Yp.x), 0.5f * (Yk.y + Yp.y) };
    float sn, csn;
    __sincosf(3.14159265358979f * (float)k / (float)FFT_N, &sn, &csn);
    cx w = { csn, sn };
    cx Op = cmul(w, Dp);
    z[__brev((unsigned)k) >> (32 - FFT_LOG2)] = { Ep.x - Op.y, Ep.y + Op.x };
    if (k > 0 && k < FFT_N / 2) {
      cx Ep2 = { 0.5f * (Yp.x + Yk.x), 0.5f * (Yp.y - Yk.y) };
      cx Dp2 = { 0.5f * (Yp.x - Yk.x), 0.5f * (Yp.y + Yk.y) };
      float sn2, cs2;
      __sincosf(3.14159265358979f * (float)(FFT_N - k) / (float)FFT_N, &sn2, &cs2);
      cx w2 = { cs2, sn2 };
      cx Op2 = cmul(w2, Dp2);
      z[__brev((unsigned)(FFT_N - k)) >> (32 - FFT_LOG2)] = { Ep2.x - Op2.y, Ep2.y + Op2.x };
    }
  }
}

// wet_ex[e*4+x] = conv2S(imp[e], res[e,x])[:S] * interp(defoF[e,x])
__global__ __launch_bounds__(1024) void k_ifft_wet(
    const cx* __restrict__ specImp, const cx* __restrict__ specRes,
    const float* __restrict__ defoF, float* __restrict__ wet_ex)
{
  extern __shared__ cx zw[];
  const int b = blockIdx.x;
  const int e = b >> 2;
  conv_spectra_to_lds(zw, specImp + (size_t)e * FFT_N, specRes + (size_t)b * FFT_N);
  __syncthreads();
  fft_lds(zw, +1.0f);
  const float inv = 1.0f / (float)FFT_N;
  const float* dF = defoF + b * FRAMES;
  const float sc = (float)FRAMES / (float)S;
  for (int n2 = threadIdx.x; n2 < S / 2; n2 += 1024) {
    int t0 = 2 * n2;
    wet_ex[(size_t)b * S + t0]     = zw[n2].x * inv * interp_row(dF, FRAMES, sc, t0);
    wet_ex[(size_t)b * S + t0 + 1] = zw[n2].y * inv * interp_row(dF, FRAMES, sc, t0 + 1);
  }
}

// final mix + dirac scheduling (conv with one-hot == delay with zero fill)
__global__ void k_sched(const float* __restrict__ imp, const float* __restrict__ wet_ex,
                        const float* __restrict__ mmix, const float* __restrict__ amplitudes,
                        const int* __restrict__ sel_time, float* __restrict__ scheduled)
{
  const int e = blockIdx.y;
  const int t = blockIdx.x * 256 + threadIdx.x;
  const int sh = sel_time[e] * HOP;
  float v = 0.0f;
  if (t >= sh) {
    const int u = t - sh;
    float wet = 0.0f;
#pragma unroll
    for (int x = 0; x < IX; ++x) wet += wet_ex[(size_t)(e * IX + x) * S + u];
    float fv = imp[(size_t)e * S + u] * mmix[e * 2] + wet * mmix[e * 2 + 1];
    v = fv * amplitudes[e];
  }
  scheduled[(size_t)e * S + t] = v;
}

// out = scheduled*rmix0 + conv2S(scheduled, room)[:S]*rmix1
__global__ __launch_bounds__(1024) void k_ifft_verb(
    const cx* __restrict__ specSched, const cx* __restrict__ specRoom,
    const float* __restrict__ scheduled, const float* __restrict__ rmix2,
    float* __restrict__ out)
{
  extern __shared__ cx zv[];
  const int e = blockIdx.x;
  conv_spectra_to_lds(zv, specSched + (size_t)e * FFT_N, specRoom + (size_t)e * FFT_N);
  __syncthreads();
  fft_lds(zv, +1.0f);
  const float inv = 1.0f / (float)FFT_N;
  const float r0 = rmix2[e * 2], r1 = rmix2[e * 2 + 1];
  for (int n2 = threadIdx.x; n2 < S / 2; n2 += 1024) {
    int t0 = 2 * n2;
    size_t o = (size_t)e * S + t0;
    out[o]     = scheduled[o]     * r0 + zv[n2].x * inv * r1;
    out[o + 1] = scheduled[o + 1] * r0 + zv[n2].y * inv * r1;
  }
}

// ---------------------------------------------------------------------------
extern "C" void kernel_launch(void* const* d_in, const int* in_sizes, int n_in,
                              void* d_out, int out_size, void* d_ws, size_t ws_size,
                              hipStream_t stream)
{
  const float* in_noise_resonance    = (const float*)d_in[0];
  const float* in_noise_deformations = (const float*)d_in[1];
  const float* in_noise_mixes        = (const float*)d_in[2];
  const float* in_envelopes          = (const float*)d_in[3];
  const float* in_resonances         = (const float*)d_in[4];
  const float* in_res_filter         = (const float*)d_in[5];
  const float* in_deformations       = (const float*)d_in[6];
  const float* in_decays             = (const float*)d_in[7];
  const float* in_mixes              = (const float*)d_in[8];
  const float* in_amplitudes         = (const float*)d_in[9];
  const float* in_times              = (const float*)d_in[10];
  const float* in_room_choice        = (const float*)d_in[11];
  const float* in_room_mix           = (const float*)d_in[12];
  const float* in_e_items            = (const float*)d_in[13];
  const float* in_n_items            = (const float*)d_in[14];
  const float* in_r_items            = (const float*)d_in[15];
  const float* in_waves              = (const float*)d_in[16];
  const float* in_d_items            = (const float*)d_in[17];
  const float* in_warp_items         = (const float*)d_in[18];
  const float* in_noise_warp_items   = (const float*)d_in[19];
  const float* in_verb_items         = (const float*)d_in[20];
  const float* in_env_noise          = (const float*)d_in[21];
  (void)in_sizes; (void)n_in; (void)out_size; (void)ws_size;

  // --- workspace layout (ints first, then floats; cx regions 8B aligned) ---
  int* wi = (int*)d_ws;
  int* sel_env  = wi;        int* sel_res  = wi + 32;   int* sel_nres = wi + 160;
  int* sel_rf   = wi + 224;  int* sel_def  = wi + 288;  int* sel_ndef = wi + 320;
  int* sel_dec  = wi + 352;  int* sel_time = wi + 384;  int* sel_room = wi + 416;

  float* fb = (float*)d_ws + 512;
  size_t off = 0;
  auto alloc = [&](size_t n) { float* p = fb + off; off += n; return p; };
  float* nmix    = alloc(64);
  float* mmix    = alloc(64);
  float* rmix2   = alloc(64);
  float* defoF   = alloc((size_t)E * IX * FRAMES);
  float* ndefF   = alloc((size_t)E * NX * FRAMES);
  float* dec128  = alloc((size_t)E * FRAMES);
  float* rfF     = alloc((size_t)E * NFS);
  float* nF      = alloc((size_t)E * NX * NFS);
  float* impulses= alloc((size_t)E * S);
  float* imp     = alloc((size_t)E * S);
  float* Abuf    = alloc((size_t)128 * 4096);
  float* res0    = alloc((size_t)128 * S);
  float* resb    = alloc((size_t)128 * S);
  float* wet_ex  = alloc((size_t)128 * S);
  float* sched   = alloc((size_t)E * S);
  cx* specImp    = (cx*)alloc((size_t)E * FFT_N * 2);     // reused for specSched
  cx* specRes    = (cx*)alloc((size_t)128 * FFT_N * 2);   // reused for specRoom
  cx* specSched  = specImp;
  cx* specRoom   = specRes;

  const size_t FFT_LDS = (size_t)FFT_N * sizeof(cx);         // 256 KB
  const size_t GEMM_LDS = (size_t)16 * 4096 * sizeof(float); // 256 KB

  // 1) selections / mixes / tables
  k_prep<<<1, 256, 0, stream>>>(
      in_envelopes, in_resonances, in_noise_resonance, in_res_filter,
      in_deformations, in_noise_deformations, in_decays, in_times,
      in_room_choice, in_noise_mixes, in_mixes, in_room_mix,
      in_d_items, in_warp_items, in_noise_warp_items, in_n_items,
      sel_env, sel_res, sel_nres, sel_rf, sel_def, sel_ndef, sel_dec,
      sel_time, sel_room, nmix, mmix, rmix2, defoF, ndefF, dec128, rfF, nF);

  // 2) impulses (envelope * env_noise, padded to S)
  k_env<<<dim3(S / 256, E), 256, 0, stream>>>(in_e_items, in_env_noise, sel_env, impulses);

  // 3) wave-bank synthesis GEMM via WMMA
  k_build_A<<<dim3(4096 / 256, 128), 256, 0, stream>>>(in_r_items, sel_res, Abuf);
  k_wave_gemm<<<dim3(2048 / 8, 8), 256, GEMM_LDS, stream>>>(Abuf, in_waves, res0);

  // 4) short FIR convolutions
  k_noise<<<dim3(S / 256, E), 256, 0, stream>>>(impulses, nF, ndefF, nmix, imp);
  k_resconv<<<dim3(S / 256, 128), 256, 0, stream>>>(res0, rfF, dec128, resb);

  // 5) long convolutions imp (*) res via in-LDS FFT
  k_fft_fwd<<<E, 1024, FFT_LDS, stream>>>(imp, (const int*)nullptr, specImp);
  k_fft_fwd<<<128, 1024, FFT_LDS, stream>>>(resb, (const int*)nullptr, specRes);
  k_ifft_wet<<<128, 1024, FFT_LDS, stream>>>(specImp, specRes, defoF, wet_ex);

  // 6) final mix + one-hot dirac scheduling (pure delay)
  k_sched<<<dim3(S / 256, E), 256, 0, stream>>>(imp, wet_ex, mmix, in_amplitudes,
                                                sel_time, sched);

  // 7) reverb conv + output mix
  k_fft_fwd<<<E, 1024, FFT_LDS, stream>>>(sched, (const int*)nullptr, specSched);
  k_fft_fwd<<<E, 1024, FFT_LDS, stream>>>(in_verb_items, sel_room, specRoom);
  k_ifft_verb<<<E, 1024, FFT_LDS, stream>>>(specSched, specRoom, sched, rmix2,
                                            (float*)d_out);
}